// VectorLogicNet_8813272891458
// MI455X (gfx1250) — compile-verified
//
#include <hip/hip_runtime.h>
#include <cstdint>

// ---------------------------------------------------------------------------
// VectorLogicNet scan: B=rows, N=64 cols, L=63 sequential weighted-power-mean
// aggregation steps per row. Log2-domain accumulator -> 1 transcendental per
// (geometric) step. Data streamed HBM -> LDS via gfx1250 async-to-LDS ops.
// ---------------------------------------------------------------------------

#define EPS_F 1e-6f
#define LOG2_EPS (-19.931568569324174f) // log2(1e-6)

constexpr int kN = 64;           // columns per row
constexpr int kL = 63;           // aggregation steps
constexpr int kWaves = 4;        // waves per block
constexpr int kThreads = kWaves * 32;
constexpr int kRowsPerWave = 32; // one row per lane
constexpr int kRowsPerBlock = kWaves * kRowsPerWave; // 128
constexpr int kStride = 68;      // padded dwords per LDS row (272B, 16B-aligned)

typedef int v4i __attribute__((ext_vector_type(4)));
typedef __attribute__((address_space(1))) v4i* g_v4i_ptr;
typedef __attribute__((address_space(3))) v4i* s_v4i_ptr;

__device__ __forceinline__ float fast_exp2(float x) {
  return __builtin_amdgcn_exp2f(x);   // v_exp_f32
}

__device__ __forceinline__ void async_copy_b128(const float* g, float* l) {
#if __has_builtin(__builtin_amdgcn_global_load_async_to_lds_b128)
  __builtin_amdgcn_global_load_async_to_lds_b128(
      (g_v4i_ptr)(uintptr_t)g, (s_v4i_ptr)(uint32_t)(uintptr_t)l, 0, 0);
#else
  asm volatile("global_load_async_to_lds_b128 %0, %1, off"
               :: "v"((uint32_t)(uintptr_t)l), "v"((uint64_t)(uintptr_t)g)
               : "memory");
#endif
}

__device__ __forceinline__ void wait_async0() {
#if __has_builtin(__builtin_amdgcn_s_wait_asynccnt)
  __builtin_amdgcn_s_wait_asynccnt(0);
#else
  asm volatile("s_wait_asynccnt 0x0" ::: "memory");
#endif
}

__global__ __launch_bounds__(kThreads)
void vln_scan_kernel(const float* __restrict__ x,
                     const float* __restrict__ a_raw,
                     const float* __restrict__ w_raw,
                     float* __restrict__ out, int B) {
  __shared__ float  tile[kWaves * kRowsPerWave * kStride]; // 34816 B
  __shared__ float4 sp[kL];                                // {w_acc, w_new, r, a}
  __shared__ float  swmin, swmax;

  const int tid  = threadIdx.x;
  const int wave = tid >> 5;
  const int lane = tid & 31;

  // ---- per-block parameter prep (63 scalars, shared by every row) ----
  if (tid == 0) {
    float mn = w_raw[0], mx = w_raw[0];
    for (int i = 1; i < kL; ++i) {
      float w = w_raw[i];
      mn = fminf(mn, w);
      mx = fmaxf(mx, w);
    }
    swmin = mn; swmax = mx;
  }
  __syncthreads();
  if (tid < kL) {
    float denom = fmaxf(swmax - swmin, 1e-8f);
    float wn = fminf(fmaxf((w_raw[tid] - swmin) / denom, 0.0f), 1.0f);
    float a  = 3.0f / (1.0f + __expf(-a_raw[tid])) - 1.0f; // sigmoid*3-1
    float ac = fminf(fmaxf(a, EPS_F), 1.0f - EPS_F);
    float r  = (1.0f - 2.0f * ac) / (ac * (1.0f - ac));
    sp[tid] = make_float4(1.0f - wn, wn, r, a);
  }

  // ---- async-copy this wave's 32-row tile into LDS (16 x b128, each 512B
  //      fully coalesced: lanes 0-15 cover row 2i, lanes 16-31 row 2i+1) ----
  const long rowBase = (long)blockIdx.x * kRowsPerBlock + wave * kRowsPerWave;
  float* ldsBase = &tile[wave * kRowsPerWave * kStride];
  const int subRow = lane >> 4;  // 0 or 1
  const int chunk  = lane & 15;  // 16B chunk index within the row
#pragma unroll
  for (int i = 0; i < 16; ++i) {
    int  rloc = 2 * i + subRow;
    long grow = rowBase + rloc;
    if (grow >= B) grow = B - 1;                 // tail clamp (dup load, store guarded)
    const float* gsrc = x + grow * kN + chunk * 4;
    float*       ldst = ldsBase + rloc * kStride + chunk * 4; // 16B aligned
    async_copy_b128(gsrc, ldst);
  }

  __syncthreads();   // params visible to all waves
  wait_async0();     // this wave's tile resident in LDS

  // ---- log2-domain scan over 63 steps; lane owns one row ----
  const float* rowp = ldsBase + lane * kStride;

  float x0  = rowp[0];
  float xc0 = fminf(fmaxf(x0, EPS_F), 1.0f);
  float lacc = (x0 != x0) ? x0 : __log2f(xc0);   // log2(clip(x0)) w/ NaN passthrough

  float zl_last = 0.0f, a_last = 0.0f;
  for (int s = 0; s < kL; ++s) {
    float4 p  = sp[s];                            // uniform broadcast (ds_load_b128)
    float  xr = rowp[s + 1];                      // ds_load_b32, 2-way conflict max
    float  xc = fminf(fmaxf(xr, EPS_F), 1.0f);
    float  lr = (xr != xr) ? xr : __log2f(xc);
    float  r  = p.z;
    float  zl;
    if (fabsf(r) < 0.001f) {
      // geometric limit: z = acc^wa * right^wn  ->  log2 z = wa*lacc + wn*lr
      zl = p.x * lacc + p.y * lr;
    } else {
      // power mean: z = (wa*acc^r + wn*right^r)^(1/r)
      float t = p.x * fast_exp2(r * lacc) + p.y * fast_exp2(r * lr);
      zl = __log2f(t) * (1.0f / r);
    }
    zl_last = zl; a_last = p.w;
    if (zl != zl) {
      // scan steps replace NaN with a_i (then next step clips); step 0 has no
      // NaN filter in the reference, so propagate there.
      lacc = (s == 0) ? zl : __log2f(fminf(fmaxf(p.w, EPS_F), 1.0f));
    } else {
      lacc = fminf(fmaxf(zl, LOG2_EPS), 0.0f);    // clip(z, eps, 1) in log domain
    }
  }

  float res = (zl_last != zl_last) ? a_last : fast_exp2(zl_last);
  long orow = rowBase + lane;
  if (orow < B) out[orow] = res;
}

extern "C" void kernel_launch(void* const* d_in, const int* in_sizes, int n_in,
                              void* d_out, int out_size, void* d_ws, size_t ws_size,
                              hipStream_t stream) {
  const float* x     = (const float*)d_in[0];
  const float* a_raw = (const float*)d_in[1];
  const float* w_raw = (const float*)d_in[2];
  float*       out   = (float*)d_out;

  const int B = in_sizes[0] / kN;
  const int grid = (B + kRowsPerBlock - 1) / kRowsPerBlock;
  hipLaunchKernelGGL(vln_scan_kernel, dim3(grid), dim3(kThreads), 0, stream,
                     x, a_raw, w_raw, out, B);
}